// DialogueGCNModel_67087389164147
// MI455X (gfx1250) — compile-verified
//
#include <hip/hip_runtime.h>
#include <math.h>

#define SEQ    300
#define BATCH  256
#define DIM    200
#define HID    128
#define NREL   8
#define NBASES 8
#define WPAST  10
#define WFUT   10
#define NCLS   6
#define NNODES (SEQ * BATCH)   // 76800
#define KMAX   200             // largest GEMM K (B panel rows staged in LDS)

typedef float v2f __attribute__((ext_vector_type(2)));
typedef float v8f __attribute__((ext_vector_type(8)));
typedef unsigned int v4u __attribute__((ext_vector_type(4)));
typedef int v4i __attribute__((ext_vector_type(4)));
typedef int v8i __attribute__((ext_vector_type(8)));

#if defined(__has_builtin)
#if __has_builtin(__builtin_amdgcn_tensor_load_to_lds)
#define USE_TDM 1
#endif
#endif

#ifdef USE_TDM
// ---------------------------------------------------------------------------
// TDM: DMA a 2D fp32 tile (rows x tile_cols) from global (row stride
// row_stride elems) into LDS at byte offset lds_off. Columns >= valid_cols
// read as zero (TDM OOB-zero) -> free edge padding. Wave-level op, EXEC
// ignored; caller issues from one wave and waits TENSORcnt.
// ---------------------------------------------------------------------------
__device__ __forceinline__ void tdm_load_2d_f32(unsigned lds_off,
                                                const float* gptr,
                                                unsigned valid_cols,
                                                unsigned tile_cols,
                                                unsigned rows,
                                                unsigned long long row_stride) {
  unsigned long long ga = (unsigned long long)(uintptr_t)gptr;
  v4u g0;
  g0.x = 1u;                                          // count=1, user descriptor
  g0.y = lds_off;                                     // lds_addr (bytes)
  g0.z = (unsigned)(ga & 0xFFFFFFFFull);              // global_addr[31:0]
  g0.w = (unsigned)((ga >> 32) & 0x1FFFFFFull) | (2u << 30);  // addr[56:32], type=2
  v8i g1;
  g1[0] = (int)(2u << 16);                            // wg_mask=0, data_size=4B
  g1[1] = (int)((valid_cols & 0xFFFFu) << 16);        // tensor_dim0[15:0] @ b63:48
  g1[2] = (int)(((valid_cols >> 16) & 0xFFFFu) | ((rows & 0xFFFFu) << 16));
  g1[3] = (int)(((rows >> 16) & 0xFFFFu) | ((tile_cols & 0xFFFFu) << 16)); // tile_dim0
  g1[4] = (int)(rows & 0xFFFFu);                      // tile_dim1=rows, tile_dim2=0
  g1[5] = (int)(row_stride & 0xFFFFFFFFull);          // tensor_dim0_stride[31:0]
  g1[6] = (int)((row_stride >> 32) & 0xFFFFull);      // stride[47:32], dim1_stride=0
  g1[7] = 0;
  v4i z4 = {0, 0, 0, 0};
#if __clang_major__ >= 23
  v8i z8 = {0, 0, 0, 0, 0, 0, 0, 0};
  __builtin_amdgcn_tensor_load_to_lds(g0, g1, z4, z4, z8, 0);
#else
  __builtin_amdgcn_tensor_load_to_lds(g0, g1, z4, z4, 0);
#endif
}
#endif  // USE_TDM

// ---------------------------------------------------------------------------
// K0: X[n, d] = features[s, b, d] with n = b*SEQ + s  (dialogue-major nodes)
// ---------------------------------------------------------------------------
__global__ __launch_bounds__(256) void k_build_x(const float* __restrict__ f,
                                                 float* __restrict__ X) {
  size_t i = (size_t)blockIdx.x * 256 + threadIdx.x;
  if (i >= (size_t)NNODES * DIM) return;
  int d = (int)(i % DIM);
  int n = (int)(i / DIM);
  int b = n / SEQ, s = n % SEQ;
  X[i] = f[((size_t)s * BATCH + b) * DIM + d];
}

// ---------------------------------------------------------------------------
// K1: wrel[r,d,h] = sum_q comp[r,q] * basis[q,d,h]
// ---------------------------------------------------------------------------
__global__ __launch_bounds__(256) void k_build_wrel(const float* __restrict__ comp,
                                                    const float* __restrict__ basis,
                                                    float* __restrict__ wrel) {
  int i = blockIdx.x * 256 + threadIdx.x;
  if (i >= NREL * DIM * HID) return;
  int dh = i % (DIM * HID);
  int r  = i / (DIM * HID);
  float acc = 0.f;
#pragma unroll
  for (int q = 0; q < NBASES; ++q)
    acc += comp[r * NBASES + q] * basis[(size_t)q * DIM * HID + dh];
  wrel[i] = acc;
}

// ---------------------------------------------------------------------------
// Generic fp32 WMMA GEMM (optionally batched over blockIdx.y):
//   C[y] = A @ B[y] (+bias)(+=C)(relu),  C[y] = C + y*strideC, B[y] = B + y*strideB
// Block = 8 waves; wave computes 16x64; block covers 128 rows x 64 cols.
// B column panel (K x 64) staged in LDS via TDM (or cooperative loads),
// zero-padded past Ncols. M % 128 == 0, K % 4 == 0, K <= KMAX.
// ---------------------------------------------------------------------------
__global__ __launch_bounds__(256) void k_gemm_wmma(
    const float* __restrict__ A, const float* __restrict__ B,
    const float* __restrict__ bias, float* __restrict__ C,
    int M, int K, int Ncols, int lda, int ldb, int ldc,
    size_t strideB, size_t strideC, int accumulate, int relu) {
  __shared__ float Bs[KMAX * 64];   // 51.2 KB

  int wave = threadIdx.x >> 5;
  int lane = threadIdx.x & 31;
  int half = lane >> 4;     // 0: lanes 0-15, 1: lanes 16-31
  int l16  = lane & 15;

  int nRowBlocks = M >> 7;
  int rowBlock = blockIdx.x % nRowBlocks;
  int colBlock = blockIdx.x / nRowBlocks;
  int m0 = (rowBlock << 7) + (wave << 4);
  int n0 = colBlock << 6;

  const float* Bp = B + (size_t)blockIdx.y * strideB;
  float* Cp = C + (size_t)blockIdx.y * strideC;

#ifdef USE_TDM
  if (threadIdx.x < 32) {
    tdm_load_2d_f32((unsigned)(uintptr_t)&Bs[0], Bp + n0,
                    (unsigned)(Ncols - n0), 64u, (unsigned)K,
                    (unsigned long long)ldb);
    __builtin_amdgcn_s_wait_tensorcnt(0);
  }
#else
  for (int idx = threadIdx.x; idx < K * 64; idx += 256) {
    int k = idx >> 6, cl = idx & 63;
    int col = n0 + cl;
    Bs[idx] = (col < Ncols) ? Bp[(size_t)k * ldb + col] : 0.f;
  }
#endif
  __syncthreads();

  v8f c[4];
#pragma unroll
  for (int t = 0; t < 4; ++t)
#pragma unroll
    for (int v = 0; v < 8; ++v) c[t][v] = 0.f;

  // A fragment (ISA 16x4 f32 layout): lane l16 owns row m0+l16;
  // lanes 0-15 read K=k0,k0+1; lanes 16-31 read K=k0+2,k0+3.
  const float* Arow = A + (size_t)(m0 + l16) * lda;
  __builtin_prefetch(Arow, 0, 3);

  for (int k0 = 0; k0 < K; k0 += 4) {
    int ka = k0 + 2 * half;
    v2f a = *(const v2f*)(Arow + ka);   // 8B-aligned: lda, ka even
#pragma unroll
    for (int t = 0; t < 4; ++t) {
      int cidx = t * 16 + l16;
      v2f b;
      b.x = Bs[ka * 64 + cidx];
      b.y = Bs[(ka + 1) * 64 + cidx];
      c[t] = __builtin_amdgcn_wmma_f32_16x16x4_f32(
          false, a, false, b, (short)0, c[t], false, false);
    }
  }

#pragma unroll
  for (int t = 0; t < 4; ++t) {
    int col = n0 + t * 16 + l16;
    if (col < Ncols) {
      float bv = bias ? bias[col] : 0.f;
#pragma unroll
      for (int v = 0; v < 8; ++v) {
        int row = m0 + (half << 3) + v;   // C layout: VGPR v -> M=v / M=8+v
        size_t idx = (size_t)row * ldc + col;
        float val = c[t][v] + bv;
        if (accumulate) val += Cp[idx];
        if (relu) val = fmaxf(val, 0.f);
        Cp[idx] = val;
      }
    }
  }
}

// ---------------------------------------------------------------------------
// K3: windowed-softmax scores.
// scale2 is (N x SEQ) row-major, N = b*SEQ + s, column i = attention slot.
// For (b,i): e_s = exp(scale2[b,s,i] - max_s); S = sum_win e + 1e-10*sum_out e;
// scores_win[(b*SEQ+i)*32 + o] = e_{i-10+o}/S for o=0..20 (0 if out of range).
// One wave per (b, 32-wide i tile); each lane owns one full column via LDS.
// ---------------------------------------------------------------------------
__global__ __launch_bounds__(32) void k_scores(const float* __restrict__ scale2,
                                               float* __restrict__ sw) {
  __shared__ float tile[SEQ * 32];   // 37.5 KB
  int b  = blockIdx.x / 10;
  int i0 = (blockIdx.x % 10) * 32;
  int lane = threadIdx.x;
  int i = i0 + lane;
  int ic = (i < SEQ) ? i : (SEQ - 1);

  for (int s = 0; s < SEQ; ++s)
    tile[s * 32 + lane] = scale2[((size_t)(b * SEQ + s)) * SEQ + ic];
  __syncthreads();

  float mx = -3.0e38f;
  for (int s = 0; s < SEQ; ++s) mx = fmaxf(mx, tile[s * 32 + lane]);

  float total = 0.f, winsum = 0.f;
  for (int s = 0; s < SEQ; ++s) {
    float e = expf(tile[s * 32 + lane] - mx);
    total += e;
    if (s >= i - WPAST && s <= i + WFUT) winsum += e;
  }
  float S = winsum + 1e-10f * (total - winsum);

  if (i < SEQ) {
    size_t base = ((size_t)(b * SEQ + i)) * 32;
    for (int o = 0; o < 32; ++o) {
      int j = i - WPAST + o;
      float val = 0.f;
      if (o < (WPAST + WFUT + 1) && j >= 0 && j < SEQ)
        val = expf(tile[j * 32 + lane] - mx) / S;
      sw[base + o] = val;
    }
  }
}

// ---------------------------------------------------------------------------
// K5: h1[t,h] = xroot[t,h](has bias_r) + sum_{s in [jt-10, jt+10]}
//              coeff(s->jt) * xw[rel(s,jt), b*SEQ+s, h]
// ---------------------------------------------------------------------------
__global__ __launch_bounds__(128) void k_h1(const float* __restrict__ xroot,
                                            const float* __restrict__ sw,
                                            const float* __restrict__ xw,
                                            const int* __restrict__ speakers,
                                            float* __restrict__ h1) {
  int t = blockIdx.x;
  int h = threadIdx.x;
  int b = t / SEQ, jt = t % SEQ;
  float acc = xroot[(size_t)t * HID + h];
  int spt = speakers[jt * BATCH + b];
  int s_lo = (jt - WFUT < 0) ? 0 : jt - WFUT;
  int s_hi = (jt + WPAST >= SEQ) ? SEQ - 1 : jt + WPAST;
  for (int s = s_lo; s <= s_hi; ++s) {
    float coeff = sw[((size_t)(b * SEQ + s)) * 32 + (jt - s + WPAST)];
    int sps = speakers[s * BATCH + b];
    int rel = sps * 4 + spt * 2 + ((s < jt) ? 0 : 1);
    acc += coeff * xw[(((size_t)rel) * NNODES + (b * SEQ + s)) * HID + h];
  }
  h1[(size_t)t * HID + h] = acc;
}

// ---------------------------------------------------------------------------
// K7: h2[t,h] = t1[t,h] + b_gc[h] + sum_{s in window(t)} t2[b*SEQ+s, h]
// ---------------------------------------------------------------------------
__global__ __launch_bounds__(128) void k_h2(const float* __restrict__ t1,
                                            const float* __restrict__ t2,
                                            const float* __restrict__ b_gc,
                                            float* __restrict__ h2) {
  int t = blockIdx.x;
  int h = threadIdx.x;
  int b = t / SEQ, jt = t % SEQ;
  float acc = t1[(size_t)t * HID + h] + b_gc[h];
  int s_lo = (jt - WFUT < 0) ? 0 : jt - WFUT;
  int s_hi = (jt + WPAST >= SEQ) ? SEQ - 1 : jt + WPAST;
  for (int s = s_lo; s <= s_hi; ++s)
    acc += t2[((size_t)(b * SEQ + s)) * HID + h];
  h2[(size_t)t * HID + h] = acc;
}

// ---------------------------------------------------------------------------
// K9: logits = hidden @ W_fc + b_fc; out = log_softmax(logits)
// ---------------------------------------------------------------------------
__global__ __launch_bounds__(256) void k_logits(const float* __restrict__ hidden,
                                                const float* __restrict__ Wfc,
                                                const float* __restrict__ bfc,
                                                float* __restrict__ out) {
  int row = blockIdx.x * 256 + threadIdx.x;
  if (row >= NNODES) return;
  float acc[NCLS];
#pragma unroll
  for (int c = 0; c < NCLS; ++c) acc[c] = bfc[c];
  const float* hr = hidden + (size_t)row * HID;
  for (int h = 0; h < HID; ++h) {
    float hv = hr[h];
#pragma unroll
    for (int c = 0; c < NCLS; ++c) acc[c] += hv * Wfc[h * NCLS + c];
  }
  float m = acc[0];
#pragma unroll
  for (int c = 1; c < NCLS; ++c) m = fmaxf(m, acc[c]);
  float sum = 0.f;
#pragma unroll
  for (int c = 0; c < NCLS; ++c) sum += expf(acc[c] - m);
  float lse = m + logf(sum);
#pragma unroll
  for (int c = 0; c < NCLS; ++c) out[(size_t)row * NCLS + c] = acc[c] - lse;
}

// ---------------------------------------------------------------------------
extern "C" void kernel_launch(void* const* d_in, const int* in_sizes, int n_in,
                              void* d_out, int out_size, void* d_ws, size_t ws_size,
                              hipStream_t stream) {
  (void)in_sizes; (void)n_in; (void)out_size; (void)ws_size;

  const float* features = (const float*)d_in[0];
  const int*   speakers = (const int*)d_in[1];
  // d_in[2] (edge_index) is implied by the fixed +/-10 window; not needed.
  const float* W_att  = (const float*)d_in[3];
  const float* b_att  = (const float*)d_in[4];
  const float* basis  = (const float*)d_in[5];
  const float* comp   = (const float*)d_in[6];
  const float* root   = (const float*)d_in[7];
  const float* bias_r = (const float*)d_in[8];
  const float* W1     = (const float*)d_in[9];
  const float* W2     = (const float*)d_in[10];
  const float* b_gc   = (const float*)d_in[11];
  const float* W_lin  = (const float*)d_in[12];
  const float* b_lin  = (const float*)d_in[13];
  const float* W_fc   = (const float*)d_in[14];
  const float* b_fc   = (const float*)d_in[15];
  float* out = (float*)d_out;

  // Workspace layout (floats), phase-overlapped big region:
  float* ws = (float*)d_ws;
  float* X      = ws;               // 15,360,000
  float* wrel   = ws + 15360000;    //    204,800
  float* sw     = ws + 15564800;    //  2,457,600
  float* xroot  = ws + 18022400;    //  9,830,400
  float* h1     = ws + 27852800;    //  9,830,400
  float* big    = ws + 37683200;    // 78,643,200 (phased)
  float* scale2 = big;                       // phase 1 (N x 300)
  float* xw     = big;                       // phase 2 (8 x N x 128)
  float* t1     = big;                       // phase 3
  float* t2     = big +  9830400;
  float* h2     = big + 19660800;
  float* hidden = big + 29491200;

  const int nRB = NNODES / 128;  // 600 row blocks

  hipLaunchKernelGGL(k_build_x, dim3(((size_t)NNODES * DIM + 255) / 256), dim3(256),
                     0, stream, features, X);
  hipLaunchKernelGGL(k_build_wrel, dim3((NREL * DIM * HID + 255) / 256), dim3(256),
                     0, stream, comp, basis, wrel);

  // scale2 = X @ W_att + b_att   (76800 x 300, K=200)
  hipLaunchKernelGGL(k_gemm_wmma, dim3(nRB * 5, 1), dim3(256), 0, stream,
                     X, W_att, b_att, scale2, NNODES, DIM, SEQ, DIM, SEQ, SEQ,
                     (size_t)0, (size_t)0, 0, 0);
  // windowed softmax scores
  hipLaunchKernelGGL(k_scores, dim3(BATCH * 10), dim3(32), 0, stream, scale2, sw);

  // xw[r] = X @ wrel[r]  -- single batched launch over blockIdx.y = r
  hipLaunchKernelGGL(k_gemm_wmma, dim3(nRB * 2, NREL), dim3(256), 0, stream,
                     X, wrel, (const float*)nullptr, xw,
                     NNODES, DIM, HID, DIM, HID, HID,
                     (size_t)DIM * HID, (size_t)NNODES * HID, 0, 0);
  // xroot = X @ root + bias_r
  hipLaunchKernelGGL(k_gemm_wmma, dim3(nRB * 2, 1), dim3(256), 0, stream,
                     X, root, bias_r, xroot, NNODES, DIM, HID, DIM, HID, HID,
                     (size_t)0, (size_t)0, 0, 0);
  // h1 = windowed RGCN aggregation + xroot
  hipLaunchKernelGGL(k_h1, dim3(NNODES), dim3(HID), 0, stream,
                     xroot, sw, xw, speakers, h1);

  // t1 = h1 @ W1 ; t2 = h1 @ W2   (overwrite xw region -- h1 done)
  hipLaunchKernelGGL(k_gemm_wmma, dim3(nRB * 2, 1), dim3(256), 0, stream,
                     h1, W1, (const float*)nullptr, t1, NNODES, HID, HID, HID, HID, HID,
                     (size_t)0, (size_t)0, 0, 0);
  hipLaunchKernelGGL(k_gemm_wmma, dim3(nRB * 2, 1), dim3(256), 0, stream,
                     h1, W2, (const float*)nullptr, t2, NNODES, HID, HID, HID, HID, HID,
                     (size_t)0, (size_t)0, 0, 0);
  // h2 = t1 + window-sum(t2) + b_gc
  hipLaunchKernelGGL(k_h2, dim3(NNODES), dim3(HID), 0, stream, t1, t2, b_gc, h2);

  // hidden = relu([X, h2] @ W_lin + b_lin), split as two GEMMs
  hipLaunchKernelGGL(k_gemm_wmma, dim3(nRB * 2, 1), dim3(256), 0, stream,
                     X, W_lin, b_lin, hidden, NNODES, DIM, HID, DIM, HID, HID,
                     (size_t)0, (size_t)0, 0, 0);
  hipLaunchKernelGGL(k_gemm_wmma, dim3(nRB * 2, 1), dim3(256), 0, stream,
                     h2, W_lin + (size_t)DIM * HID, (const float*)nullptr, hidden,
                     NNODES, HID, HID, HID, HID, HID,
                     (size_t)0, (size_t)0, 1, 1);

  // logits + log_softmax
  hipLaunchKernelGGL(k_logits, dim3((NNODES + 255) / 256), dim3(256), 0, stream,
                     hidden, W_fc, b_fc, out);
}